// BronxLayer_6201932775820
// MI455X (gfx1250) — compile-verified
//
#include <hip/hip_runtime.h>
#include <math.h>

// BRONX layer for MI455X (gfx1250): fp32 WMMA (V_WMMA_F32_16X16X4_F32) pipeline.
// N=4096 nodes, F=256 features. All fp32.
// Big GEMMs register-blocked: one wave computes a 32x64 output tile
// (2 A-fragments x 4 B-fragments -> 8 WMMAs per K=4 step).
// K loop tiled by one 128B cache line (32 floats): one WGP-scope
// global_prefetch_b8 per stream per line, 64 WMMAs per outer iteration.

#define NNODE 4096
#define FDIM  256

typedef __attribute__((ext_vector_type(2))) float v2f;
typedef __attribute__((ext_vector_type(8))) float v8f;

__device__ __forceinline__ v8f wmma_f32(v2f a, v2f b, v8f c) {
  // D = A(16x4 f32) * B(4x16 f32) + C(16x16 f32)
  return __builtin_amdgcn_wmma_f32_16x16x4_f32(
      /*neg_a=*/false, a, /*neg_b=*/false, b,
      /*c_mod=*/(short)0, c, /*reuse_a=*/false, /*reuse_b=*/false);
}

// WGP-scope prefetch (ISA scope 0: pull into all cache levels, stop on WGP hit).
__device__ __forceinline__ void prefetch_wgp(const float* p) {
  asm volatile("global_prefetch_b8 %0, off" :: "v"(p));
}

// ---------------------------------------------------------------------------
// K1: per-row LayerNorm(h) and L1-normalize(x).  grid = N rows, block = 256.
// ---------------------------------------------------------------------------
__global__ void bronx_norm(const float* __restrict__ h, const float* __restrict__ x,
                           const float* __restrict__ gamma, const float* __restrict__ beta,
                           float* __restrict__ hn, float* __restrict__ xn) {
  __shared__ float red[256];
  const int row = blockIdx.x, t = threadIdx.x;
  const float hv = h[(size_t)row * FDIM + t];
  const float xv = x[(size_t)row * FDIM + t];

  red[t] = hv; __syncthreads();
  for (int s = 128; s > 0; s >>= 1) { if (t < s) red[t] += red[t + s]; __syncthreads(); }
  const float mu = red[0] * (1.0f / FDIM);
  __syncthreads();

  const float d = hv - mu;
  red[t] = d * d; __syncthreads();
  for (int s = 128; s > 0; s >>= 1) { if (t < s) red[t] += red[t + s]; __syncthreads(); }
  const float inv = rsqrtf(red[0] * (1.0f / FDIM) + 1e-5f);
  __syncthreads();

  red[t] = fabsf(xv); __syncthreads();
  for (int s = 128; s > 0; s >>= 1) { if (t < s) red[t] += red[t + s]; __syncthreads(); }
  const float l1 = fmaxf(red[0], 1e-12f);

  hn[(size_t)row * FDIM + t] = d * inv * gamma[t] + beta[t];
  xn[(size_t)row * FDIM + t] = xv / l1;
}

// ---------------------------------------------------------------------------
// NT GEMM  C[m,n] = scale * sum_k A[m,k]*B[n,k]   (K multiple of 32)
// One 32x64 output tile per wave: 2 A-frags x 4 B-frags -> 8 WMMAs / K-step.
// Used for: k/q projections, S_h logits, S_x affinity.
// ---------------------------------------------------------------------------
__global__ void bronx_gemm_nt(const float* __restrict__ A, const float* __restrict__ B,
                              float* __restrict__ C, int M, int Ncol, int K,
                              int lda, int ldb, int ldc, float scale) {
  const int wave = (blockIdx.x * blockDim.x + threadIdx.x) >> 5;
  const int lane = threadIdx.x & 31;
  const int tilesN = Ncol >> 6;                         // 64-wide column tiles
  const int wm = wave / tilesN, wn = wave - wm * tilesN;
  if (wm >= (M >> 5)) return;                           // wave-uniform
  const int idx = lane & 15;
  const int khalf = (lane >> 4) << 1;                   // 0 or 2
  const float* a0 = A + (size_t)(wm * 32 + idx) * lda + khalf;
  const float* a1 = a0 + (size_t)16 * lda;
  const float* b0 = B + (size_t)(wn * 64 + idx) * ldb + khalf;
  v8f acc[8] = {};                                      // [mi*4 + ni]
  for (int kb = 0; kb < K; kb += 32) {                  // one 128B line per iter
    if (kb + 32 < K) {                                  // next-line WGP prefetch
      prefetch_wgp(a0 + kb + 32);                       // (skip last: stay in row)
      prefetch_wgp(b0 + kb + 32);
    }
#pragma unroll
    for (int ks = 0; ks < 32; ks += 4) {
      const int k = kb + ks;
      v2f a[2], b[4];
      a[0].x = a0[k]; a[0].y = a0[k + 1];
      a[1].x = a1[k]; a[1].y = a1[k + 1];
#pragma unroll
      for (int ni = 0; ni < 4; ++ni) {
        const float* bp = b0 + (size_t)(ni * 16) * ldb;
        b[ni].x = bp[k]; b[ni].y = bp[k + 1];
      }
#pragma unroll
      for (int mi = 0; mi < 2; ++mi)
#pragma unroll
        for (int ni = 0; ni < 4; ++ni)
          acc[mi * 4 + ni] = wmma_f32(a[mi], b[ni], acc[mi * 4 + ni]);
    }
  }
  const int r0 = wm * 32 + ((lane >> 4) << 3);
#pragma unroll
  for (int mi = 0; mi < 2; ++mi)
#pragma unroll
    for (int ni = 0; ni < 4; ++ni) {
      const int col = wn * 64 + ni * 16 + idx;
#pragma unroll
      for (int g = 0; g < 8; ++g)
        C[(size_t)(r0 + mi * 16 + g) * ldc + col] = acc[mi * 4 + ni][g] * scale;
    }
}

// ---------------------------------------------------------------------------
// K4: row softmax of S (4096 cols) in place.  grid = N rows, block = 256.
// ---------------------------------------------------------------------------
__global__ void bronx_softmax(float* __restrict__ S) {
  __shared__ float red[256];
  const int t = threadIdx.x;
  float* r = S + (size_t)blockIdx.x * NNODE;
  float mx = -3.402823466e38f;
  for (int j = t; j < NNODE; j += 256) mx = fmaxf(mx, r[j]);
  red[t] = mx; __syncthreads();
  for (int s = 128; s > 0; s >>= 1) { if (t < s) red[t] = fmaxf(red[t], red[t + s]); __syncthreads(); }
  mx = red[0]; __syncthreads();
  float sum = 0.0f;
  for (int j = t; j < NNODE; j += 256) { float e = expf(r[j] - mx); r[j] = e; sum += e; }
  red[t] = sum; __syncthreads();
  for (int s = 128; s > 0; s >>= 1) { if (t < s) red[t] += red[t + s]; __syncthreads(); }
  const float inv = 1.0f / red[0];
  __syncthreads();
  for (int j = t; j < NNODE; j += 256) r[j] *= inv;
}

// ---------------------------------------------------------------------------
// K5: per-row stats of a_x: [diag, sum, std(ddof=1)] -> st[N,3]
// ---------------------------------------------------------------------------
__global__ void bronx_stats(const float* __restrict__ Sx, float* __restrict__ st) {
  __shared__ float red[256];
  const int row = blockIdx.x, t = threadIdx.x;
  const float* r = Sx + (size_t)row * NNODE;
  float sum = 0.0f;
  for (int j = t; j < NNODE; j += 256) sum += r[j];
  red[t] = sum; __syncthreads();
  for (int s = 128; s > 0; s >>= 1) { if (t < s) red[t] += red[t + s]; __syncthreads(); }
  const float total = red[0];
  const float mean = total * (1.0f / NNODE);
  __syncthreads();
  float ss = 0.0f;
  for (int j = t; j < NNODE; j += 256) { float d = r[j] - mean; ss += d * d; }
  red[t] = ss; __syncthreads();
  for (int s = 128; s > 0; s >>= 1) { if (t < s) red[t] += red[t + s]; __syncthreads(); }
  if (t == 0) {
    st[row * 3 + 0] = r[row];
    st[row * 3 + 1] = total;
    st[row * 3 + 2] = sqrtf(red[0] * (1.0f / (NNODE - 1)));
  }
}

// ---------------------------------------------------------------------------
// K6: x_out[j,f] = sum_i (m00*Sx[j,i] + m10*Sh[j,i]) * xn[i,f] + x0[j,f]
// NN GEMM over K=4096; mixing folded into A fragments; 32x64 tile per wave.
// ---------------------------------------------------------------------------
__global__ void bronx_xout(const float* __restrict__ Sx, const float* __restrict__ Sh,
                           const float* __restrict__ xn, const float* __restrict__ x0,
                           const float* __restrict__ mixing, float* __restrict__ out) {
  const float e0 = expf(mixing[0]), e1 = expf(mixing[2]);   // column 0 of mixing
  const float m00 = e0 / (e0 + e1), m10 = e1 / (e0 + e1);
  const int wave = (blockIdx.x * blockDim.x + threadIdx.x) >> 5;
  const int lane = threadIdx.x & 31;
  const int wm = wave >> 2, wn = wave & 3;                  // FDIM/64 = 4 col tiles
  const int idx = lane & 15;
  const int khalf = (lane >> 4) << 1;
  const float* ax0 = Sx + (size_t)(wm * 32 + idx) * NNODE + khalf;
  const float* ah0 = Sh + (size_t)(wm * 32 + idx) * NNODE + khalf;
  const float* ax1 = ax0 + (size_t)16 * NNODE;
  const float* ah1 = ah0 + (size_t)16 * NNODE;
  const int c0 = wn * 64 + idx;
  v8f acc[8] = {};
  for (int kb = 0; kb < NNODE; kb += 32) {
    if (kb + 32 < NNODE) {
      prefetch_wgp(ax0 + kb + 32);
      prefetch_wgp(ah0 + kb + 32);
    }
#pragma unroll
    for (int ks = 0; ks < 32; ks += 4) {
      const int k = kb + ks;
      v2f a[2], b[4];
      a[0].x = m00 * ax0[k]     + m10 * ah0[k];
      a[0].y = m00 * ax0[k + 1] + m10 * ah0[k + 1];
      a[1].x = m00 * ax1[k]     + m10 * ah1[k];
      a[1].y = m00 * ax1[k + 1] + m10 * ah1[k + 1];
      const float* brow0 = xn + (size_t)(k + khalf) * FDIM + c0;
#pragma unroll
      for (int ni = 0; ni < 4; ++ni) {
        b[ni].x = brow0[ni * 16];
        b[ni].y = brow0[ni * 16 + FDIM];
      }
#pragma unroll
      for (int mi = 0; mi < 2; ++mi)
#pragma unroll
        for (int ni = 0; ni < 4; ++ni)
          acc[mi * 4 + ni] = wmma_f32(a[mi], b[ni], acc[mi * 4 + ni]);
    }
  }
  const int r0 = wm * 32 + ((lane >> 4) << 3);
#pragma unroll
  for (int mi = 0; mi < 2; ++mi)
#pragma unroll
    for (int ni = 0; ni < 4; ++ni) {
      const int col = c0 + ni * 16;
#pragma unroll
      for (int g = 0; g < 8; ++g) {
        const size_t o = (size_t)(r0 + mi * 16 + g) * FDIM + col;
        out[o] = acc[mi * 4 + ni][g] + x0[o];
      }
    }
}

// ---------------------------------------------------------------------------
// K7: h_agg[i,f] = sum_j (m01*Sx[j,i] + m11*Sh[j,i]) * hn[j,f]   (TN GEMM)
// A accessed column-wise -> lane-contiguous (coalesced); 32x64 tile per wave.
// ---------------------------------------------------------------------------
__global__ void bronx_hagg(const float* __restrict__ Sx, const float* __restrict__ Sh,
                           const float* __restrict__ hn, const float* __restrict__ mixing,
                           float* __restrict__ hagg) {
  const float e0 = expf(mixing[1]), e1 = expf(mixing[3]);   // column 1 of mixing
  const float m01 = e0 / (e0 + e1), m11 = e1 / (e0 + e1);
  const int wave = (blockIdx.x * blockDim.x + threadIdx.x) >> 5;
  const int lane = threadIdx.x & 31;
  const int wm = wave >> 2, wn = wave & 3;
  const int idx = lane & 15;
  const int khalf = (lane >> 4) << 1;
  const int rg0 = wm * 32 + idx;                            // A^T row == A column
  const int c0 = wn * 64 + idx;
  v8f acc[8] = {};
  for (int kb = 0; kb < NNODE; kb += 4) {
    const size_t k0 = (size_t)(kb + khalf) * NNODE + rg0;
    v2f a[2], b[4];
    a[0].x = m01 * Sx[k0]              + m11 * Sh[k0];
    a[0].y = m01 * Sx[k0 + NNODE]      + m11 * Sh[k0 + NNODE];
    a[1].x = m01 * Sx[k0 + 16]         + m11 * Sh[k0 + 16];
    a[1].y = m01 * Sx[k0 + NNODE + 16] + m11 * Sh[k0 + NNODE + 16];
    const float* brow0 = hn + (size_t)(kb + khalf) * FDIM + c0;
#pragma unroll
    for (int ni = 0; ni < 4; ++ni) {
      b[ni].x = brow0[ni * 16];
      b[ni].y = brow0[ni * 16 + FDIM];
    }
#pragma unroll
    for (int mi = 0; mi < 2; ++mi)
#pragma unroll
      for (int ni = 0; ni < 4; ++ni)
        acc[mi * 4 + ni] = wmma_f32(a[mi], b[ni], acc[mi * 4 + ni]);
  }
  const int r0 = wm * 32 + ((lane >> 4) << 3);
#pragma unroll
  for (int mi = 0; mi < 2; ++mi)
#pragma unroll
    for (int ni = 0; ni < 4; ++ni) {
      const int col = c0 + ni * 16;
#pragma unroll
      for (int g = 0; g < 8; ++g)
        hagg[(size_t)(r0 + mi * 16 + g) * FDIM + col] = acc[mi * 4 + ni][g];
    }
}

// ---------------------------------------------------------------------------
// K8: h_out = elu(cat([h_agg, i]) @ w_v.T) + h0.  w_v is [256, 259] row-major.
// WMMA over K=256, then 3 stat columns in the epilogue.  16x16 tile per wave.
// ---------------------------------------------------------------------------
__global__ void bronx_hout(const float* __restrict__ hagg, const float* __restrict__ st,
                           const float* __restrict__ wv, const float* __restrict__ h0,
                           float* __restrict__ out) {
  const int wave = (blockIdx.x * blockDim.x + threadIdx.x) >> 5;
  const int lane = threadIdx.x & 31;
  const int tm = wave >> 4, tn = wave & 15;
  const int idx = lane & 15;
  const int khalf = (lane >> 4) << 1;
  const float* arow = hagg + (size_t)(tm * 16 + idx) * FDIM + khalf;
  const float* brow = wv + (size_t)(tn * 16 + idx) * 259 + khalf;
  v8f acc = {};
  for (int kb = 0; kb < FDIM; kb += 4) {
    v2f a, b;
    a.x = arow[kb]; a.y = arow[kb + 1];
    b.x = brow[kb]; b.y = brow[kb + 1];
    acc = wmma_f32(a, b, acc);
  }
  const int row = tm * 16 + ((lane >> 4) << 3);
  const int col = tn * 16 + idx;
  const float wv0 = wv[(size_t)col * 259 + 256];
  const float wv1 = wv[(size_t)col * 259 + 257];
  const float wv2 = wv[(size_t)col * 259 + 258];
#pragma unroll
  for (int g = 0; g < 8; ++g) {
    const int r = row + g;
    float v = acc[g] + st[r * 3 + 0] * wv0 + st[r * 3 + 1] * wv1 + st[r * 3 + 2] * wv2;
    v = (v > 0.0f) ? v : (expf(v) - 1.0f);                  // ELU, alpha=1
    out[(size_t)r * FDIM + col] = v + h0[(size_t)r * FDIM + col];
  }
}

// ---------------------------------------------------------------------------
extern "C" void kernel_launch(void* const* d_in, const int* in_sizes, int n_in,
                              void* d_out, int out_size, void* d_ws, size_t ws_size,
                              hipStream_t stream) {
  const float* h      = (const float*)d_in[0];
  const float* x      = (const float*)d_in[1];
  const float* w_k    = (const float*)d_in[2];
  const float* w_q    = (const float*)d_in[3];
  const float* w_v    = (const float*)d_in[4];
  const float* mixing = (const float*)d_in[5];
  const float* gamma  = (const float*)d_in[6];
  const float* beta   = (const float*)d_in[7];

  float* out_h = (float*)d_out;
  float* out_x = out_h + (size_t)NNODE * FDIM;

  const size_t NF = (size_t)NNODE * FDIM;
  float* ws = (float*)d_ws;
  float* hn    = ws; ws += NF;
  float* xn    = ws; ws += NF;
  float* kbuf  = ws; ws += NF;
  float* qbuf  = ws; ws += NF;
  float* hagg  = ws; ws += NF;
  float* Sh    = ws; ws += (size_t)NNODE * NNODE;
  float* Sx    = ws; ws += (size_t)NNODE * NNODE;
  float* stats = ws;

  // 1) row normalizations
  bronx_norm<<<NNODE, 256, 0, stream>>>(h, x, gamma, beta, hn, xn);
  // 2) projections k = hn @ w_k^T, q = hn @ w_q^T
  //    wave-tasks = (4096/32)*(256/64) = 512 -> 64 blocks of 8 waves
  bronx_gemm_nt<<<64, 256, 0, stream>>>(hn, w_k, kbuf, NNODE, FDIM, FDIM, FDIM, FDIM, FDIM, 1.0f);
  bronx_gemm_nt<<<64, 256, 0, stream>>>(hn, w_q, qbuf, NNODE, FDIM, FDIM, FDIM, FDIM, FDIM, 1.0f);
  // 3) logits S_h = (k q^T)/16, affinity S_x = xn xn^T
  //    wave-tasks = (4096/32)*(4096/64) = 8192 -> 1024 blocks
  bronx_gemm_nt<<<1024, 256, 0, stream>>>(kbuf, qbuf, Sh, NNODE, NNODE, FDIM, FDIM, FDIM, NNODE, 0.0625f);
  bronx_gemm_nt<<<1024, 256, 0, stream>>>(xn, xn, Sx, NNODE, NNODE, FDIM, FDIM, FDIM, NNODE, 1.0f);
  // 4) softmax rows of S_h in place
  bronx_softmax<<<NNODE, 256, 0, stream>>>(Sh);
  // 5) row stats of S_x (before mixing)
  bronx_stats<<<NNODE, 256, 0, stream>>>(Sx, stats);
  // 6) x_out = a_x2 @ xn + x0   (512 wave-tasks -> 64 blocks)
  bronx_xout<<<64, 256, 0, stream>>>(Sx, Sh, xn, x, mixing, out_x);
  // 7) h_agg = a_h2^T @ hn      (512 wave-tasks -> 64 blocks)
  bronx_hagg<<<64, 256, 0, stream>>>(Sx, Sh, hn, mixing, hagg);
  // 8) h_out = elu(cat(h_agg, i) @ w_v^T) + h0  (4096 tile-waves -> 512 blocks)
  bronx_hout<<<512, 256, 0, stream>>>(hagg, stats, w_v, h, out_h);
}